// LBFGSUpdate_21844203667769
// MI455X (gfx1250) — compile-verified
//
#include <hip/hip_runtime.h>
#include <hip/hip_bf16.h>

// L-BFGS two-loop recursion, restructured as:
//   1) gram_kernel:    G = Z'*Z'^T with Z' = [S(20 rows); Y(20 rows); g; pad..48]
//                      using V_WMMA_F32_16X16X4_F32 (6 C-tiles cover S^TY, Y^TY, S^Tg, Y^Tg)
//   2) reduce_recurse: deterministic cross-block reduction + O(H^2) scalar two-loop
//                      recursion producing gamma, coefS[20], coefY[20]
//   3) combine_kernel: out = gamma*g + sum coefS[m]*S[m] + sum coefY[m]*Y[m]

#define HLEN 20
#define NTILE 6
#define TILE_ELEMS 256
#define BLK_ENTRIES (NTILE * TILE_ELEMS)   // 1536 floats per block partial
#define GRAM_BLOCKS 256
#define GRAM_THREADS 256
#define GRAM_WAVES (GRAM_THREADS / 32)

typedef float v2f __attribute__((ext_vector_type(2)));
typedef float v4f __attribute__((ext_vector_type(4)));
typedef float v8f __attribute__((ext_vector_type(8)));

// ---------------------------------------------------------------------------
// Kernel 1: Gram pass. Z' is 48 x D (rows 0..19 = S, 20..39 = Y, 40 = g,
// 41..47 = dummy -> garbage lands only in C rows/cols >= 41, which are never
// read, so no masking needed). Each wave owns a contiguous K-chunk; per K-step
// of 4 it loads one float2 per lane per 16-row group (landing exactly in the
// 16x4 f32 A-matrix WMMA layout: lanes 0-15 hold K=k..k+1, lanes 16-31 hold
// K=k+2..k+3) and issues 6 v_wmma_f32_16x16x4_f32. For a Gram product the
// B operand (4x16, KxN) has the identical register image as A (16x4, MxK).
// Tiles (r,c) in {0,1,2} x {1,2} cover all of G[0..47][16..47], containing
//   S^TY = G[0:20][20:40], Y^TY = G[20:40][20:40],
//   S^Tg = G[0:20][40],    Y^Tg = G[20:40][40].
// ---------------------------------------------------------------------------
__global__ __launch_bounds__(GRAM_THREADS) void gram_kernel(
    const float* __restrict__ g, const float* __restrict__ S,
    const float* __restrict__ Y, float* __restrict__ P, int D) {
  __shared__ float red[GRAM_WAVES][BLK_ENTRIES];   // 48 KB

  const int tid  = threadIdx.x;
  const int lane = tid & 31;
  const int wib  = tid >> 5;
  const int totalWaves = gridDim.x * GRAM_WAVES;
  // Wave-uniform id in an SGPR so loop bounds are scalar (no exec-mask loop).
  const int waveId = __builtin_amdgcn_readfirstlane(blockIdx.x * GRAM_WAVES + wib);

  // Per-lane row base pointers for the three 16-row groups.
  const float* ptr[3];
#pragma unroll
  for (int r = 0; r < 3; ++r) {
    const int row = 16 * r + (lane & 15);
    const float* base;
    if (row < HLEN)          base = S + (size_t)row * (size_t)D;
    else if (row < 2 * HLEN) base = Y + (size_t)(row - HLEN) * (size_t)D;
    else                     base = g;            // row 40 = g; rows 41..47 dummy
    ptr[r] = base + 2 * (lane >> 4);              // K sub-offset per WMMA f32 A layout
  }

  // Contiguous K chunk per wave (keeps cacheline reuse across k-steps).
  int chunk = ((D + totalWaves - 1) / totalWaves + 3) & ~3;
  int kBeg = waveId * chunk;
  int kEnd = kBeg + chunk;
  if (kEnd > D) kEnd = D;

  v8f acc[NTILE] = {};

#pragma unroll 2
  for (int k = kBeg; k < kEnd; k += 4) {
    v2f a0 = __builtin_nontemporal_load((const v2f*)(ptr[0] + k));
    v2f a1 = __builtin_nontemporal_load((const v2f*)(ptr[1] + k));
    v2f a2 = __builtin_nontemporal_load((const v2f*)(ptr[2] + k));
    acc[0] = __builtin_amdgcn_wmma_f32_16x16x4_f32(false, a0, false, a1, (short)0, acc[0], false, false);
    acc[1] = __builtin_amdgcn_wmma_f32_16x16x4_f32(false, a0, false, a2, (short)0, acc[1], false, false);
    acc[2] = __builtin_amdgcn_wmma_f32_16x16x4_f32(false, a1, false, a1, (short)0, acc[2], false, false);
    acc[3] = __builtin_amdgcn_wmma_f32_16x16x4_f32(false, a1, false, a2, (short)0, acc[3], false, false);
    acc[4] = __builtin_amdgcn_wmma_f32_16x16x4_f32(false, a2, false, a1, (short)0, acc[4], false, false);
    acc[5] = __builtin_amdgcn_wmma_f32_16x16x4_f32(false, a2, false, a2, (short)0, acc[5], false, false);
  }

  // Deterministic in-block reduction: each wave writes its own LDS slice,
  // then every thread folds the 8 wave slices in a fixed order.
#pragma unroll
  for (int t = 0; t < NTILE; ++t)
#pragma unroll
    for (int v = 0; v < 8; ++v)
      red[wib][t * TILE_ELEMS + v * 32 + lane] = acc[t][v];
  __syncthreads();

  for (int e = tid; e < BLK_ENTRIES; e += blockDim.x) {
    float s = red[0][e];
#pragma unroll
    for (int w = 1; w < GRAM_WAVES; ++w) s += red[w][e];
    P[(size_t)blockIdx.x * BLK_ENTRIES + e] = s;
  }
}

// ---------------------------------------------------------------------------
// Kernel 2: fold the per-block partials into G[48][48] (deterministic order),
// then one thread runs the two-loop recursion on the tiny Gram system.
//   backward: a[i] = rho[p_i]*(S^Tg[p_i] - sum_{j>i} a[j]*SY[p_i][p_j])
//   gamma    = SY[l][l]/YY[l][l]  (l = p_0), guarded
//   forward : beta_i = rho[p_i]*(gamma*(Yg[p_i] - sum_j a[j]*YY[p_i][p_j])
//                                + sum_{q<i} c[q]*SY[p_q][p_i]),  c[i]=a[i]-beta_i
//   out coefs: gamma, coefS[p_i]=c[i], coefY[p_i]=-gamma*a[i]
// ---------------------------------------------------------------------------
#define SYm(m, n) G[(m) * 48 + 20 + (n)]
#define YYm(m, n) G[(20 + (m)) * 48 + 20 + (n)]
#define Sgm(m)    G[(m) * 48 + 40]
#define Ygm(m)    G[(20 + (m)) * 48 + 40]

__global__ __launch_bounds__(256) void reduce_recurse_kernel(
    const float* __restrict__ P, int nblocks,
    const float* __restrict__ rho, const int* __restrict__ isp,
    float* __restrict__ G, float* __restrict__ CO) {
  const int tid = threadIdx.x;

  // Phase 1: reduce partials and scatter C-tile elements into G.
  for (int e = tid; e < BLK_ENTRIES; e += blockDim.x) {
    float s = 0.0f;
    for (int b = 0; b < nblocks; ++b) s += P[(size_t)b * BLK_ENTRIES + e];
    const int t = e >> 8, idx = e & 255;
    const int v = idx >> 5, l = idx & 31;
    const int r = t >> 1, c = (t & 1) + 1;
    const int M = 16 * r + v + 8 * (l >> 4);   // WMMA f32 C layout
    const int N = 16 * c + (l & 15);
    G[M * 48 + N] = s;
  }
  __syncthreads();

  if (tid != 0) return;

  const int is = isp[0];
  int p[HLEN];
  for (int i = 0; i < HLEN; ++i) p[i] = ((i + is) % HLEN + HLEN) % HLEN;

  float a[HLEN], c[HLEN];
  for (int i = HLEN - 1; i >= 0; --i) {              // backward loop
    float s = Sgm(p[i]);
    for (int j = i + 1; j < HLEN; ++j) s -= a[j] * SYm(p[i], p[j]);
    a[i] = rho[p[i]] * s;
  }
  const int l0 = p[0];
  const float den = YYm(l0, l0);
  const float gamma = (den > 1e-10f) ? (SYm(l0, l0) / den) : 1.0f;

  for (int i = 0; i < HLEN; ++i) {                   // forward loop
    float s = Ygm(p[i]);
    for (int j = 0; j < HLEN; ++j) s -= a[j] * YYm(p[i], p[j]);
    s *= gamma;
    for (int q = 0; q < i; ++q) s += c[q] * SYm(p[q], p[i]);  // y_{p_i}.s_{p_q}
    const float beta = rho[p[i]] * s;
    c[i] = a[i] - beta;
  }

  CO[0] = gamma;
  for (int i = 0; i < HLEN; ++i) CO[1 + p[i]]        = c[i];
  for (int i = 0; i < HLEN; ++i) CO[1 + HLEN + p[i]] = -gamma * a[i];
}

// ---------------------------------------------------------------------------
// Kernel 3: streaming linear combination, b128 loads, NT hints on the
// g, S and Y loads and the out store (every stream is single-use).
// ---------------------------------------------------------------------------
__global__ __launch_bounds__(256) void combine_kernel(
    const float* __restrict__ g, const float* __restrict__ S,
    const float* __restrict__ Y, const float* __restrict__ CO,
    float* __restrict__ out, int D) {
  const float gamma = CO[0];
  float cs[HLEN], cy[HLEN];
#pragma unroll
  for (int m = 0; m < HLEN; ++m) { cs[m] = CO[1 + m]; cy[m] = CO[1 + HLEN + m]; }

  const int D4 = D >> 2;
  const int stride = gridDim.x * blockDim.x;
  for (int i = blockIdx.x * blockDim.x + threadIdx.x; i < D4; i += stride) {
    const size_t o = (size_t)i * 4;
    v4f acc = __builtin_nontemporal_load((const v4f*)(g + o));
    acc *= gamma;
#pragma unroll
    for (int m = 0; m < HLEN; ++m) {
      v4f sv = __builtin_nontemporal_load((const v4f*)(S + (size_t)m * D + o));
      acc += cs[m] * sv;
    }
#pragma unroll
    for (int m = 0; m < HLEN; ++m) {
      v4f yv = __builtin_nontemporal_load((const v4f*)(Y + (size_t)m * D + o));
      acc += cy[m] * yv;
    }
    __builtin_nontemporal_store(acc, (v4f*)(out + o));
  }

  if (blockIdx.x == 0 && threadIdx.x == 0) {         // scalar tail (D % 4)
    for (int t = D4 * 4; t < D; ++t) {
      float accv = gamma * g[t];
      for (int m = 0; m < HLEN; ++m) accv = fmaf(cs[m], S[(size_t)m * D + t], accv);
      for (int m = 0; m < HLEN; ++m) accv = fmaf(cy[m], Y[(size_t)m * D + t], accv);
      out[t] = accv;
    }
  }
}

// ---------------------------------------------------------------------------
extern "C" void kernel_launch(void* const* d_in, const int* in_sizes, int n_in,
                              void* d_out, int out_size, void* d_ws, size_t ws_size,
                              hipStream_t stream) {
  (void)n_in; (void)out_size; (void)ws_size;
  const float* g   = (const float*)d_in[0];   // pytree
  const float* S   = (const float*)d_in[1];   // y_diff_history  (s_k)
  const float* Y   = (const float*)d_in[2];   // grad_diff_history (y_k)
  const float* rho = (const float*)d_in[3];   // inner_history
  const int*   isp = (const int*)d_in[4];     // index_start
  float* out = (float*)d_out;
  float* W   = (float*)d_ws;
  const int D = in_sizes[0];

  // Workspace layout (floats): partials | G(48x48) | coefs(1+40)  (~1.51 MB)
  float* P  = W;
  float* G  = W + (size_t)GRAM_BLOCKS * BLK_ENTRIES;
  float* CO = G + 48 * 48;

  gram_kernel<<<GRAM_BLOCKS, GRAM_THREADS, 0, stream>>>(g, S, Y, P, D);
  reduce_recurse_kernel<<<1, 256, 0, stream>>>(P, GRAM_BLOCKS, rho, isp, G, CO);

  int D4 = D >> 2;
  int cBlocks = (D4 + 255) / 256;
  if (cBlocks > 4096) cBlocks = 4096;
  if (cBlocks < 1) cBlocks = 1;
  combine_kernel<<<cBlocks, 256, 0, stream>>>(g, S, Y, CO, out, D);
}